// TopK_26594437496962
// MI455X (gfx1250) — compile-verified
//
#include <hip/hip_runtime.h>
#include <stdint.h>

// ------------------------------------------------------------------
// TopK(K=256) per row + ReLU + scatter-back for x[8192][16384] fp32.
//
// Roofline: dense in/out -> 512MB read + 512MB write ~= 1 GiB.
// At 23.3 TB/s HBM: ~46 us floor. No matmul anywhere, so WMMA has no
// role; the CDNA5 features that matter are the async global->LDS path
// (ASYNCcnt + s_wait_asynccnt) and wave32 LDS atomics for radix select.
//
// One 512-thread (16 wave32) workgroup per row:
//  1. async-stage the 64KB row into LDS (overlapped with hist zeroing)
//  2. keys -> VGPRs (order-preserving uint32 map), 32 per thread
//  3. 4x 8-bit MSB-first radix select w/ per-wave privatized LDS
//     histograms (stride 257 to stagger the 64 banks)
//  4. scatter pass: >T keep relu(x), ==T claim tie slots, else 0;
//     coalesced b128 stores.
// ------------------------------------------------------------------

#define NCOLS   16384
#define KSEL    256
#define TPB     512
#define EPT     32              // elements per thread
#define CHUNK   8               // EPT / 4 (float4 chunks)
#define NWAVE   (TPB / 32)      // wave32 on gfx1250
#define HSTRIDE 257             // 256 bins + 1 pad word -> stagger banks

__device__ __forceinline__ uint32_t f2k(float f) {
  // order-preserving float -> uint32 (ascending)
  uint32_t b = __float_as_uint(f);
  return (b & 0x80000000u) ? ~b : (b | 0x80000000u);
}
__device__ __forceinline__ float k2f(uint32_t k) {
  uint32_t b = (k & 0x80000000u) ? (k & 0x7FFFFFFFu) : ~k;
  return __uint_as_float(b);
}

__global__ __launch_bounds__(TPB) void topk_relu_scatter(
    const float* __restrict__ x, float* __restrict__ out) {
  __shared__ __align__(16) float lds_x[NCOLS];   // 64 KB staged row
  __shared__ uint32_t hist[NWAVE * HSTRIDE];     // per-wave private hists
  __shared__ uint32_t tot[256];                  // reduced histogram
  __shared__ uint32_t ctrl[4];                   // 0:prefix 1:kk 2:tie-counter

  const int tid = threadIdx.x;
  const uint64_t rowbase =
      (uint64_t)(uintptr_t)(x + (size_t)blockIdx.x * NCOLS);

  // ---- 1. async stage: global -> LDS (CDNA5 ASYNCcnt path) ----
  const uint32_t ldsbase = (uint32_t)(uintptr_t)&lds_x[0];
  #pragma unroll
  for (int c = 0; c < CHUNK; ++c) {
    const uint32_t off  = (uint32_t)((c * (TPB * 4) + tid * 4) * 4); // bytes
    const uint32_t ldso = ldsbase + off;
    asm volatile("global_load_async_to_lds_b128 %0, %1, %2"
                 :: "v"(ldso), "v"(off), "s"(rowbase)
                 : "memory");
  }
  // overlap with the async copy: zero histograms + init control words
  for (int i = tid; i < NWAVE * HSTRIDE; i += TPB) hist[i] = 0u;
  if (tid == 0) { ctrl[0] = 0u; ctrl[1] = KSEL; ctrl[2] = 0u; }
  asm volatile("s_wait_asynccnt 0" ::: "memory");
  __syncthreads();

  // ---- 2. pull this thread's 32 elements into VGPRs as sortable keys ----
  uint32_t key[EPT];
  #pragma unroll
  for (int c = 0; c < CHUNK; ++c) {
    const float4 v = *(const float4*)&lds_x[c * (TPB * 4) + tid * 4];
    key[c * 4 + 0] = f2k(v.x);
    key[c * 4 + 1] = f2k(v.y);
    key[c * 4 + 2] = f2k(v.z);
    key[c * 4 + 3] = f2k(v.w);
  }

  // ---- 3. MSB-first radix select (4 rounds x 8 bits) ----
  uint32_t* myhist = &hist[(tid >> 5) * HSTRIDE];   // wave32 -> tid/32
  #pragma unroll
  for (int r = 0; r < 4; ++r) {
    const uint32_t prefix = ctrl[0];
    const int shift = 24 - 8 * r;
    #pragma unroll
    for (int e = 0; e < EPT; ++e) {
      const uint32_t k = key[e];
      if ((r == 0) || ((k >> (shift + 8)) == prefix))
        atomicAdd(&myhist[(k >> shift) & 0xFFu], 1u);   // ds_add_u32
    }
    __syncthreads();
    if (tid < 256) {          // reduce 16 private copies -> tot[bin]
      uint32_t s = 0;
      #pragma unroll
      for (int w = 0; w < NWAVE; ++w) s += hist[w * HSTRIDE + tid];
      tot[tid] = s;
    }
    __syncthreads();
    // re-zero private hists for the next round (concurrent with the scan,
    // which only reads tot[])
    for (int i = tid; i < NWAVE * HSTRIDE; i += TPB) hist[i] = 0u;
    if (tid == 0) {
      const uint32_t kk = ctrl[1];
      uint32_t cum = 0, c = 0, digit = 0;
      for (int d = 255; d >= 0; --d) {
        c = tot[d];
        cum += c;
        if (cum >= kk) { digit = (uint32_t)d; break; }
      }
      ctrl[0] = (ctrl[0] << 8) | digit;   // extend prefix
      ctrl[1] = kk - (cum - c);           // rank within the tied bin
    }
    __syncthreads();
  }

  // ---- 4. scatter: key>T keep relu, key==T claim tie slot, else 0 ----
  const uint32_t T   = ctrl[0];   // full 32-bit threshold key
  const uint32_t kkf = ctrl[1];   // number of ==T elements to keep
  float* orow = out + (size_t)blockIdx.x * NCOLS;
  #pragma unroll
  for (int c = 0; c < CHUNK; ++c) {
    float4 o;
    float* op = (float*)&o;
    #pragma unroll
    for (int j = 0; j < 4; ++j) {
      const uint32_t k = key[c * 4 + j];
      float v = 0.0f;
      if (k > T) {
        v = fmaxf(k2f(k), 0.0f);
      } else if (k == T && atomicAdd(&ctrl[2], 1u) < kkf) {  // rare ties
        v = fmaxf(k2f(k), 0.0f);
      }
      op[j] = v;
    }
    *(float4*)&orow[c * (TPB * 4) + tid * 4] = o;   // global_store_b128
  }
}

extern "C" void kernel_launch(void* const* d_in, const int* in_sizes, int n_in,
                              void* d_out, int out_size, void* d_ws, size_t ws_size,
                              hipStream_t stream) {
  const float* x = (const float*)d_in[0];
  float* out = (float*)d_out;
  const int rows = in_sizes[0] / NCOLS;   // 8192 rows of 16384 fp32
  topk_relu_scatter<<<rows, TPB, 0, stream>>>(x, out);
}